// SemiMarkovCRFHead_3178275799190
// MI455X (gfx1250) — compile-verified
//
#include <hip/hip_runtime.h>
#include <hip/hip_bf16.h>

typedef __attribute__((ext_vector_type(2))) float v2f;
typedef __attribute__((ext_vector_type(8))) float v8f;

#define B_   16
#define T_   8192
#define H_   512
#define C_   24
#define K_   100
#define NEGV (-1e30f)
#define WAVES 8
#define WPAD 516   // LDS row stride for transposed W: 516%64==4 -> conflict-free b64 reads

// ---------------------------------------------------------------------------
// Kernel 1: scores[(b*T+t), c] = sum_h hidden[(b*T+t), h] * W[h, c] + bias[c]
// HBM-bound (256 MB in @ 23.3 TB/s ~= 11 us). V_WMMA_F32_16X16X4_F32.
// W_proj (48 KB, kernel-invariant) is staged ONCE per block into LDS,
// transposed to Wt[n][k] and zero-padded to n=32, so the inner loop is
// 1 global_load_b64 (A) + 2 ds_load_b64 (B frags) + 2 WMMA — no divergence,
// no per-iteration address math (immediate offsets via unroll).
//
// f32 A 16x4 layout: lanes 0-15 hold M=lane,{K0,K1}; lanes 16-31 M=lane-16,
// {K2,K3}. B 4x16 mirrors with N in lanes. C/D: VGPR v -> M=v (+8 for upper
// half-wave), N = lane&15.
// ---------------------------------------------------------------------------
__global__ __launch_bounds__(256) void smcrf_proj_wmma(
    const float* __restrict__ hidden, const float* __restrict__ W,
    const float* __restrict__ bias, float* __restrict__ scores)
{
  __shared__ float wt[32 * WPAD];                // 66 KB: Wt[n][k], n>=24 zero

  const int tid  = threadIdx.x;
  const int lane = tid & 31;
  const int wv   = tid >> 5;
  const int half = lane >> 4;                    // 0: {K0,K1}, 1: {K2,K3}
  const int rm   = lane & 15;                    // A row / B col within tile

  // one-time staging of W (transposed + padded); 48 KB, L2-resident
  for (int i = tid; i < 32 * H_; i += 256) {
    const int n = i >> 9, k = i & (H_ - 1);
    wt[n * WPAD + k] = (n < C_) ? W[k * C_ + n] : 0.f;
  }
  __syncthreads();

  const int tile = blockIdx.x * 8 + wv;          // 1024 blocks * 8 waves = 8192 tiles
  const long row0 = (long)tile * 16;
  const int kk_off = 2 * half;
  const float* Arow = hidden + (row0 + rm) * H_ + kk_off;
  const float* Bt0  = &wt[rm * WPAD + kk_off];         // N-tile 0 (cols 0..15)
  const float* Bt1  = &wt[(16 + rm) * WPAD + kk_off];  // N-tile 1 (cols 16..31)

  v8f acc0 = {0.f,0.f,0.f,0.f,0.f,0.f,0.f,0.f};
  v8f acc1 = {0.f,0.f,0.f,0.f,0.f,0.f,0.f,0.f};

  #pragma unroll 16
  for (int k0 = 0; k0 < H_; k0 += 4) {
    if ((k0 & 63) == 0) __builtin_prefetch(Arow + k0 + 256, 0, 0); // global_prefetch_b8
    v2f a  = *(const v2f*)(Arow + k0);           // 8B aligned (kk even)
    v2f b0 = *(const v2f*)(Bt0 + k0);            // ds_load_b64, conflict-free
    v2f b1 = *(const v2f*)(Bt1 + k0);
    acc0 = __builtin_amdgcn_wmma_f32_16x16x4_f32(false, a, false, b0, (short)0, acc0, false, false);
    acc1 = __builtin_amdgcn_wmma_f32_16x16x4_f32(false, a, false, b1, (short)0, acc1, false, false);
  }

  #pragma unroll
  for (int v = 0; v < 8; ++v) {
    const long r = row0 + v + 8 * half;
    scores[r * C_ + rm] = acc0[v] + bias[rm];
    if (rm < C_ - 16)                            // only cols 16..23 valid in tile 1
      scores[r * C_ + 16 + rm] = acc1[v] + bias[16 + rm];
  }
}

// ---------------------------------------------------------------------------
// Kernel 2: semi-CRF forward scan, one workgroup per batch (16 WGs, 8 waves).
// Ring buffer R[s mod K][c] = m_s - cum_s in LDS fuses the cumsum.
//   alpha_t[c] = cum_t[c] + LSE_{d<K}( R[(t-1-d)%K][c] + dur[d][c] )
//   m_t[c']    = LSE_c( alpha_t[c] + trans[c][c'] )   (via exp(trans) matmul)
//   R[t%K][c'] = m_t[c'] - cum_t[c']
// K-loop split over 8 waves as streaming (max,sum) partials; wave 0 combines
// and does the transition in one region (intra-wave LDS is in-order, so no
// barrier between the alpha_s write and its cross-lane read). 2 barriers/step.
// ---------------------------------------------------------------------------
__global__ __launch_bounds__(256) void smcrf_scan(
    const float* __restrict__ scores, const int* __restrict__ lengths,
    const float* __restrict__ trans, const float* __restrict__ dur,
    float* __restrict__ out)
{
  __shared__ float R[K_ * C_];
  __shared__ float durs[K_ * C_];
  __shared__ float eT[C_ * C_];
  __shared__ float pm[WAVES * C_];
  __shared__ float ps[WAVES * C_];
  __shared__ float alpha_s[C_];

  const int b    = blockIdx.x;
  const int tid  = threadIdx.x;
  const int lane = tid & 31;
  const int w    = tid >> 5;
  const int len  = lengths[b];
  const int c    = lane;
  const bool act = (c < C_);

  for (int i = tid; i < K_ * C_; i += 256) {
    durs[i] = dur[i];
    R[i] = (i < C_) ? 0.f : NEGV;      // slot 0 <-> s=0: m_0 - cum_0 = 0
  }
  for (int i = tid; i < C_ * C_; i += 256) eT[i] = expf(trans[i]);
  __syncthreads();

  float cum = 0.f;                     // wave 0 lanes track cum_t[c]
  const float* srow = scores + (long)b * T_ * C_;

  for (int t = 1; t <= T_; ++t) {
    // --- stage 1: per-wave partial streaming LSE over d = w, w+8, ... ---
    if (act) {
      float mx = -3.0e38f, sm = 0.f;
      int slot = (t - 1 - w) % K_; if (slot < 0) slot += K_;
      for (int d = w; d < K_; d += WAVES) {
        const float v  = R[slot * C_ + c] + durs[d * C_ + c];
        const float nm = fmaxf(mx, v);
        sm = sm * expf(mx - nm) + expf(v - nm);
        mx = nm;
        slot -= WAVES; if (slot < 0) slot += K_;
      }
      pm[w * C_ + c] = mx;
      ps[w * C_ + c] = sm;
      if (w == 0) cum += srow[(t - 1) * C_ + c];
    }
    __syncthreads();

    // --- stage 2: wave 0 combines -> alpha_t, transition LSE, ring write ---
    if (w == 0 && act) {
      float Mx = -3.0e38f;
      #pragma unroll
      for (int w2 = 0; w2 < WAVES; ++w2) Mx = fmaxf(Mx, pm[w2 * C_ + c]);
      float Sm = 0.f;
      #pragma unroll
      for (int w2 = 0; w2 < WAVES; ++w2) Sm += ps[w2 * C_ + c] * expf(pm[w2 * C_ + c] - Mx);
      const float alpha = cum + Mx + logf(Sm);
      alpha_s[c] = alpha;              // intra-wave LDS: in-order, visible below

      float am = -3.0e38f;
      #pragma unroll
      for (int cc = 0; cc < C_; ++cc) am = fmaxf(am, alpha_s[cc]);
      float s = 0.f;
      #pragma unroll
      for (int cc = 0; cc < C_; ++cc) s += expf(alpha_s[cc] - am) * eT[cc * C_ + c];
      R[(t % K_) * C_ + c] = (am + logf(s)) - cum;

      if (lane == 0 && t == len) {     // partition for this batch
        float sy = 0.f;
        for (int cc = 0; cc < C_; ++cc) sy += expf(alpha_s[cc] - am);
        out[b] = am + logf(sy);
      }
    }
    __syncthreads();                   // publishes R for next step's stage 1
  }
}

extern "C" void kernel_launch(void* const* d_in, const int* in_sizes, int n_in,
                              void* d_out, int out_size, void* d_ws, size_t ws_size,
                              hipStream_t stream) {
  const float* hidden = (const float*)d_in[0];   // (16, 8192, 512) f32
  const int*   lens   = (const int*)  d_in[1];   // (16,) i32
  const float* W      = (const float*)d_in[2];   // (512, 24) f32
  const float* bias   = (const float*)d_in[3];   // (24,) f32
  const float* trans  = (const float*)d_in[4];   // (24, 24) f32
  const float* dur    = (const float*)d_in[5];   // (100, 24) f32
  float* out    = (float*)d_out;                 // (16,) f32
  float* scores = (float*)d_ws;                  // (16*8192*24) f32 = 12.6 MB scratch

  smcrf_proj_wmma<<<dim3(1024), dim3(256), 0, stream>>>(hidden, W, bias, scores);
  smcrf_scan    <<<dim3(B_),   dim3(256), 0, stream>>>(scores, lens, trans, dur, out);
}